// FixedRadiusNeighborQuery_86277303042226
// MI455X (gfx1250) — compile-verified
//
#include <hip/hip_runtime.h>

// Fixed-radius neighbor query, CDNA5/gfx1250.
//  - d2 tiles via V_WMMA_F32_16X16X4_F32 (256 pair dot products / instruction)
//  - points staged to LDS with gfx1250 async-to-LDS loads (ASYNCcnt), double buffered
//  - ballot-compacted candidate push (one 2-lane ds_add_rtn_u32 per C-VGPR)
//  - per-query top-32 min-extraction, ascending d2 like jax.lax.top_k

#define FRNN_LIMIT   32
#define FRNN_RADIUS  0.3f
#define FRNN_CAP     96                        // per-query candidate capacity
#define FRNN_WAVES   4                         // waves per workgroup
#define FRNN_TILE_Q  16                        // queries per wave (WMMA M)
#define FRNN_CHUNK   512                       // points per async-staged chunk
#define FRNN_THREADS (FRNN_WAVES * 32)
#define FRNN_CHUNK_F (FRNN_CHUNK * 3)          // floats per chunk (6 KB)
#define FRNN_LPT     (FRNN_CHUNK_F / FRNN_THREADS) // async b32 loads/thread/chunk = 12

static_assert(FRNN_LPT == 12, "s_wait_asynccnt immediates assume 12 loads/thread");

typedef __attribute__((ext_vector_type(2))) float v2f;
typedef __attribute__((ext_vector_type(8))) float v8f;

// Issue one chunk of async global->LDS copies (per-lane b32, GV addressing).
__device__ __forceinline__ void async_stage_chunk(
    const float* __restrict__ pts_f, long fbase, long fend, long fsafe,
    unsigned lds_base, int tid)
{
#pragma unroll
    for (int r = 0; r < FRNN_LPT; ++r) {
        const int k = r * FRNN_THREADS + tid;
        long g = fbase + k;
        if (g >= fend) g = fsafe;              // clamp: consumer masks tail lanes anyway
        const unsigned lds = lds_base + (unsigned)k * 4u;
        const unsigned long long ga = (unsigned long long)(const void*)(pts_f + g);
        asm volatile("global_load_async_to_lds_b32 %0, %1, off"
                     :: "v"(lds), "v"(ga) : "memory");
    }
}

__global__ __launch_bounds__(FRNN_THREADS)
void FixedRadiusNeighborQuery_86277303042226_kernel(
    const float* __restrict__ pts,      // (N,3)
    const int*   __restrict__ prs,      // (B+1) point row splits
    int n_rs,
    const float* __restrict__ qp,       // (Q,3)
    const int*   __restrict__ qrs,      // (B+1) query row splits
    int Q,
    int*   __restrict__ idx_out,        // (Q,LIMIT)
    float* __restrict__ dist_out)       // (Q,LIMIT)
{
    __shared__ float    s_stage[2][FRNN_CHUNK_F];           // 12 KB double buffer
    __shared__ unsigned s_cnt[FRNN_WAVES][FRNN_TILE_Q];
    __shared__ float    s_d2 [FRNN_WAVES][FRNN_TILE_Q][FRNN_CAP];   // 24 KB
    __shared__ int      s_idx[FRNN_WAVES][FRNN_TILE_Q][FRNN_CAP];   // 24 KB

    const int tid  = threadIdx.x;
    const int lane = tid & 31;
    const int w    = tid >> 5;
    const int col  = lane & 15;
    const int half = lane >> 4;

    if (lane < FRNN_TILE_Q) s_cnt[w][lane] = 0u;

    // ---- block-uniform batch: all 4 tiles (64 consecutive queries) share it ----
    const int q0_block = blockIdx.x * (FRNN_WAVES * FRNN_TILE_Q);
    int b = 0;
    while (b + 2 < n_rs && qrs[b + 1] <= q0_block) ++b;
    const int pstart = prs[b];
    const int pend   = prs[b + 1];

    const int  tile        = blockIdx.x * FRNN_WAVES + w;   // wave-uniform
    const int  q0          = tile * FRNN_TILE_Q;
    const bool wave_active = (q0 < Q);

    // ---- per-wave query tile: WMMA A operand (16x4 f32 layout) + row norms ----
    v2f   a; a.x = 0.f; a.y = 0.f;
    float q2row[8];
    if (wave_active) {
        int qi = q0 + col; if (qi >= Q) qi = Q - 1;
        const float qx = qp[3 * qi + 0];
        const float qy = qp[3 * qi + 1];
        const float qz = qp[3 * qi + 2];
        const float q2 = fmaf(qx, qx, fmaf(qy, qy, qz * qz));
        a.x = half ? qz : qx;   // K0 / K2
        a.y = half ? 0.f : qy;  // K1 / K3 (pad)
#pragma unroll
        for (int v = 0; v < 8; ++v)
            q2row[v] = __shfl(q2, v + 8 * half, 32);
    }

    const unsigned lds_s0 = (unsigned)(unsigned long long)(const void*)&s_stage[0][0];
    const unsigned lds_s1 = (unsigned)(unsigned long long)(const void*)&s_stage[1][0];

    const long  fend  = (long)pend * 3;
    const long  fsafe = (long)pstart * 3;
    const int   nchunks = (pend - pstart + FRNN_CHUNK - 1) / FRNN_CHUNK;
    const float r2 = FRNN_RADIUS * FRNN_RADIUS;

    if (nchunks > 0)
        async_stage_chunk(pts, (long)pstart * 3, fend, fsafe, lds_s0, tid);

    for (int ci = 0; ci < nchunks; ++ci) {
        const int cbase = pstart + ci * FRNN_CHUNK;

        if (ci + 1 < nchunks) {
            async_stage_chunk(pts, (long)(cbase + FRNN_CHUNK) * 3, fend, fsafe,
                              (ci & 1) ? lds_s0 : lds_s1, tid);
            asm volatile("s_wait_asynccnt 12" ::: "memory"); // current chunk complete
        } else {
            asm volatile("s_wait_asynccnt 0" ::: "memory");
        }
        __syncthreads();                                     // data visible block-wide

        if (wave_active) {
            int csz = pend - cbase; if (csz > FRNN_CHUNK) csz = FRNN_CHUNK;
            const float* sp = &s_stage[ci & 1][0];

            for (int t = 0; t < csz; t += 16) {              // uniform trip count
                const int  pi     = cbase + t + col;
                const bool pvalid = (t + col) < csz;
                const int  l      = pvalid ? (t + col) : 0;

                float px = sp[3 * l + 0];
                float py = sp[3 * l + 1];
                float pz = sp[3 * l + 2];
                if (!pvalid) { px = 1e18f; py = 0.f; pz = 0.f; } // force out of radius

                const float p2 = fmaf(px, px, fmaf(py, py, pz * pz));
                v2f bv;
                bv.x = half ? pz : px;
                bv.y = half ? 0.f : py;

                v8f acc = {};
                acc = __builtin_amdgcn_wmma_f32_16x16x4_f32(
                    false, a, false, bv, (short)0, acc, false, false);

#pragma unroll
                for (int v = 0; v < 8; ++v) {
                    float d2 = fmaf(-2.0f, acc[v], q2row[v] + p2);
                    d2 = fmaxf(d2, 0.0f);
                    const bool hit = (d2 <= r2);

                    // ballot-compacted push: rows v (lanes 0-15) and v+8 (lanes 16-31)
                    const unsigned m    = (unsigned)__ballot(hit);
                    const unsigned grp  = half ? (m >> 16) : (m & 0xffffu);
                    const int      cg   = __popc(grp);
                    unsigned base = 0;
                    if (col == 0 && cg)                       // lanes 0 & 16 only
                        base = atomicAdd(&s_cnt[w][v + 8 * half], (unsigned)cg);
                    base = (unsigned)__shfl((int)base, half << 4, 32);
                    if (hit) {
                        const unsigned off = base + (unsigned)__popc(grp & ((1u << col) - 1u));
                        if (off < FRNN_CAP) {
                            const int row = v + 8 * half;
                            s_d2 [w][row][off] = d2;
                            s_idx[w][row][off] = pi;
                        }
                    }
                }
            }
        }
        __syncthreads();                                     // buffer reuse safe
    }
    __syncthreads();

    // ---- per-query top-32 extraction (ascending d2, like top_k) ----
    if (wave_active && lane < FRNN_TILE_Q) {
        const int row = lane;
        const int qi  = q0 + row;
        if (qi < Q) {
            int n = (int)s_cnt[w][row];
            if (n > FRNN_CAP) n = FRNN_CAP;
            const size_t outb = (size_t)qi * FRNN_LIMIT;
#pragma unroll 1
            for (int j = 0; j < FRNN_LIMIT; ++j) {
                float best = 1e30f;
                int   bslot = -1;
#pragma unroll 1
                for (int k = 0; k < n; ++k) {
                    const float dd = s_d2[w][row][k];
                    if (dd < best) { best = dd; bslot = k; }
                }
                if (bslot >= 0) {
                    idx_out [outb + j] = s_idx[w][row][bslot];
                    dist_out[outb + j] = sqrtf(fmaxf(best, 1e-12f));
                    s_d2[w][row][bslot] = 1e30f;             // consume
                } else {
                    idx_out [outb + j] = -1;
                    dist_out[outb + j] = 0.0f;
                }
            }
        }
    }
}

extern "C" void kernel_launch(void* const* d_in, const int* in_sizes, int n_in,
                              void* d_out, int out_size, void* d_ws, size_t ws_size,
                              hipStream_t stream) {
    const float* pts  = (const float*)d_in[0];   // points (N,3) f32
    const int*   prs  = (const int*)  d_in[1];   // row_splits (B+1) i32
    const float* qp   = (const float*)d_in[2];   // output_points (Q,3) f32
    const int*   qrs  = (const int*)  d_in[3];   // output_row_splits (B+1) i32

    const int n_rs = in_sizes[1];
    const int Q    = in_sizes[2] / 3;

    // outputs concatenated flat: idx (Q*LIMIT i32), dist (Q*LIMIT f32)
    int*   idx_out  = (int*)d_out;
    float* dist_out = (float*)d_out + (size_t)Q * FRNN_LIMIT;

    const int ntiles = (Q + FRNN_TILE_Q - 1) / FRNN_TILE_Q;
    const int blocks = (ntiles + FRNN_WAVES - 1) / FRNN_WAVES;

    FixedRadiusNeighborQuery_86277303042226_kernel
        <<<dim3(blocks), dim3(FRNN_THREADS), 0, stream>>>(
            pts, prs, n_rs, qp, qrs, Q, idx_out, dist_out);
}